// LinearAttention_7035156430915
// MI455X (gfx1250) — compile-verified
//
#include <hip/hip_runtime.h>
#include <math.h>

// ---------------------------------------------------------------------------
// MI455X (gfx1250) fp32 linear attention.
// Matrix math on V_WMMA_F32_16X16X4_F32 (fp32 in, fp32 acc) to match the fp32
// reference; GEMM tiles DMA'd straight into LDS with the CDNA5 async engine
// (GLOBAL_LOAD_ASYNC_TO_LDS_B128 / ASYNCcnt) with LDS double buffering.
// wave32; 256-thread blocks = 8 waves.
// ---------------------------------------------------------------------------

typedef __attribute__((ext_vector_type(2))) float v2f;
typedef __attribute__((ext_vector_type(8))) float v8f;

#define WMMA_F32X4(a, b, c) \
  __builtin_amdgcn_wmma_f32_16x16x4_f32(false, (a), false, (b), (short)0, (c), false, false)

// Problem constants
#define BATCH 16
#define CDIM 512
#define NDIM 1024          // 32*32 spatial
#define HEADS 8
#define HDIM 64            // CDIM / HEADS
#define QKVROWS 1536       // 3*CDIM

// --- CDNA5 async global->LDS copy (16B per lane), tracked by ASYNCcnt ------
__device__ __forceinline__ void async_copy_b128(unsigned lds_off, const void* gptr) {
  asm volatile("global_load_async_to_lds_b128 %0, %1, off"
               :: "v"(lds_off), "v"(gptr)
               : "memory");
}
__device__ __forceinline__ void wait_async0() {
  asm volatile("s_wait_asynccnt 0x0" ::: "memory");
}
__device__ __forceinline__ unsigned lds_addr_of(const void* p) {
  // LDS aperture: flat addr[31:0] is the wave-relative LDS byte address.
  return (unsigned)(unsigned long long)p;
}

// ---------------------------------------------------------------------------
// Generic GEMM:  Y[batch, M, N] = W[M, K] @ X[batch, K, N]  (+ bias[M])
// WG tile 128x128, 8 waves, wave tile 64x32 (4x2 wmma tiles), KT = 16,
// double-buffered LDS filled by async DMA.
// ---------------------------------------------------------------------------
__global__ __launch_bounds__(256) void gemm_wmma_f32_kernel(
    const float* __restrict__ W, const float* __restrict__ X,
    const float* __restrict__ bias, float* __restrict__ Y,
    int M, int K, int N) {
  const int batch = blockIdx.z;
  const int m0 = blockIdx.y * 128;
  const int n0 = blockIdx.x * 128;
  const float* __restrict__ Xb = X + (size_t)batch * K * N;
  float* __restrict__ Yb = Y + (size_t)batch * M * N;

  // [m][k] padded to 20 (80B = 5*16B rows); [k][n] padded to 132 (528B = 33*16B)
  __shared__ alignas(16) float As[2][128][20];
  __shared__ alignas(16) float Bs[2][16][132];

  const int tid = threadIdx.x;
  const int lane = tid & 31;
  const int wv = tid >> 5;           // 0..7
  const int wM = wv >> 2;            // 0..1 -> +64 in M
  const int wN = wv & 3;             // 0..3 -> +32 in N
  const int ml = lane & 15;
  const int g = lane >> 4;           // 0/1 (K-half select for fp32 fragments)

  v8f acc[4][2] = {};

  const int NT = K / 16;

  // Issue DMA for tile 0 into buffer 0.
  {
#pragma unroll
    for (int c = tid; c < 512; c += 256) {       // A: 128 rows x 4 chunks
      const int row = c >> 2, koff = (c & 3) * 4;
      async_copy_b128(lds_addr_of(&As[0][row][koff]),
                      W + (size_t)(m0 + row) * K + koff);
    }
#pragma unroll
    for (int c = tid; c < 512; c += 256) {       // B: 16 rows x 32 chunks
      const int row = c >> 5, coff = (c & 31) * 4;
      async_copy_b128(lds_addr_of(&Bs[0][row][coff]),
                      Xb + (size_t)row * N + n0 + coff);
    }
  }

  for (int t = 0; t < NT; ++t) {
    wait_async0();          // our async writes for tile t have landed
    __syncthreads();        // everyone's have; previous buffer reads all done

    if (t + 1 < NT) {       // DMA tile t+1 into the other buffer, overlapped
      const int nb = (t + 1) & 1;
      const int k1 = (t + 1) * 16;
#pragma unroll
      for (int c = tid; c < 512; c += 256) {
        const int row = c >> 2, koff = (c & 3) * 4;
        async_copy_b128(lds_addr_of(&As[nb][row][koff]),
                        W + (size_t)(m0 + row) * K + k1 + koff);
      }
#pragma unroll
      for (int c = tid; c < 512; c += 256) {
        const int row = c >> 5, coff = (c & 31) * 4;
        async_copy_b128(lds_addr_of(&Bs[nb][row][coff]),
                        Xb + (size_t)(k1 + row) * N + n0 + coff);
      }
    }

    const float (*A)[20] = As[t & 1];
    const float (*B)[132] = Bs[t & 1];

    // Batch all fragment loads for this 16-wide K tile, then the WMMA burst.
    v2f afr[4][4];   // [kk4][mi]
    v2f bfr[4][2];   // [kk4][ni]
#pragma unroll
    for (int kk4 = 0; kk4 < 4; ++kk4) {
      const int kk = kk4 * 4;
#pragma unroll
      for (int mi = 0; mi < 4; ++mi) {
        const int r = wM * 64 + mi * 16 + ml;
        afr[kk4][mi].x = A[r][kk + 2 * g];
        afr[kk4][mi].y = A[r][kk + 2 * g + 1];
      }
#pragma unroll
      for (int ni = 0; ni < 2; ++ni) {
        const int cN = wN * 32 + ni * 16 + ml;
        bfr[kk4][ni].x = B[kk + 2 * g][cN];
        bfr[kk4][ni].y = B[kk + 2 * g + 1][cN];
      }
    }
#pragma unroll
    for (int kk4 = 0; kk4 < 4; ++kk4)
#pragma unroll
      for (int mi = 0; mi < 4; ++mi)
#pragma unroll
        for (int ni = 0; ni < 2; ++ni)
          acc[mi][ni] = WMMA_F32X4(afr[kk4][mi], bfr[kk4][ni], acc[mi][ni]);
  }

  // Store: C/D layout -> vgpr e holds row (e + 8*g), col = ml
#pragma unroll
  for (int mi = 0; mi < 4; ++mi) {
#pragma unroll
    for (int ni = 0; ni < 2; ++ni) {
      const int ncol = n0 + wN * 32 + ni * 16 + ml;
#pragma unroll
      for (int e = 0; e < 8; ++e) {
        const int mrow = m0 + wM * 64 + mi * 16 + e + 8 * g;
        const float bv = bias ? bias[mrow] : 0.0f;
        Yb[(size_t)mrow * N + ncol] = acc[mi][ni][e] + bv;
      }
    }
  }
}

// ---------------------------------------------------------------------------
// k softmax: over n (row of 1024), in place inside qkv buffer.
// One 256-thread block per (b, ch) row of the K section.
// ---------------------------------------------------------------------------
__global__ __launch_bounds__(256) void softmax_k_kernel(float* __restrict__ qkv) {
  const int b = blockIdx.x >> 9;       // / 512
  const int ch = blockIdx.x & 511;
  float* row = qkv + ((size_t)b * QKVROWS + CDIM + ch) * NDIM;

  const int tid = threadIdx.x;
  const int lane = tid & 31;
  const int wv = tid >> 5;
  __shared__ float sred[8];

  float v[4];
  float m = -3.402823466e38f;
#pragma unroll
  for (int i = 0; i < 4; ++i) {
    v[i] = row[tid + i * 256];
    m = fmaxf(m, v[i]);
  }
#pragma unroll
  for (int off = 16; off > 0; off >>= 1) m = fmaxf(m, __shfl_xor(m, off, 32));
  if (lane == 0) sred[wv] = m;
  __syncthreads();
  float mAll = sred[0];
#pragma unroll
  for (int i = 1; i < 8; ++i) mAll = fmaxf(mAll, sred[i]);
  __syncthreads();

  float s = 0.0f;
#pragma unroll
  for (int i = 0; i < 4; ++i) {
    v[i] = expf(v[i] - mAll);
    s += v[i];
  }
#pragma unroll
  for (int off = 16; off > 0; off >>= 1) s += __shfl_xor(s, off, 32);
  if (lane == 0) sred[wv] = s;
  __syncthreads();
  float sAll = 0.0f;
#pragma unroll
  for (int i = 0; i < 8; ++i) sAll += sred[i];
  const float inv = 1.0f / sAll;
#pragma unroll
  for (int i = 0; i < 4; ++i) row[tid + i * 256] = v[i] * inv;
}

// ---------------------------------------------------------------------------
// q softmax: over the 64 head channels (axis=-2), in place.
// One thread per (b, h, n); adjacent threads -> adjacent n -> coalesced.
// ---------------------------------------------------------------------------
__global__ __launch_bounds__(256) void softmax_q_kernel(float* __restrict__ qkv) {
  const int idx = blockIdx.x * 256 + threadIdx.x;  // 0 .. 131071
  const int nn = idx & (NDIM - 1);
  const int bh = idx >> 10;                        // 0..127
  const int b = bh >> 3;
  const int h = bh & 7;
  float* base = qkv + ((size_t)b * QKVROWS + h * HDIM) * NDIM + nn;

  float vals[HDIM];
  float m = -3.402823466e38f;
#pragma unroll
  for (int i = 0; i < HDIM; ++i) {
    vals[i] = base[(size_t)i * NDIM];
    m = fmaxf(m, vals[i]);
  }
  float s = 0.0f;
#pragma unroll
  for (int i = 0; i < HDIM; ++i) {
    vals[i] = expf(vals[i] - m);
    s += vals[i];
  }
  const float inv = 1.0f / s;
#pragma unroll
  for (int i = 0; i < HDIM; ++i) base[(size_t)i * NDIM] = vals[i] * inv;
}

// ---------------------------------------------------------------------------
// qv[b,h] = q[b,h] (64 x 1024) @ v[b,h]^T (1024 x 64)   -> 64 x 64
// One workgroup per (b,h). q and v tiles staged in LDS (v read transposed).
// Wave w computes tiles (tm = w>>1, tn = (w&1)*2 + {0,1}).
// ---------------------------------------------------------------------------
__global__ __launch_bounds__(256) void qv_wmma_kernel(const float* __restrict__ qkv,
                                                      float* __restrict__ qvout) {
  const int bh = blockIdx.x;          // 0..127
  const int b = bh >> 3;
  const int h = bh & 7;
  const float* __restrict__ qp = qkv + ((size_t)b * QKVROWS + h * HDIM) * NDIM;
  const float* __restrict__ vp = qkv + ((size_t)b * QKVROWS + 2 * CDIM + h * HDIM) * NDIM;
  float* __restrict__ qvb = qvout + (size_t)bh * HDIM * HDIM;

  __shared__ alignas(16) float Qs[64][36];        // [c][t], KT=32, padded
  __shared__ alignas(16) float Vs[64][36];        // [d][t]

  const int tid = threadIdx.x;
  const int lane = tid & 31;
  const int wv = tid >> 5;
  const int tm = wv >> 1;             // 0..3
  const int tn2 = (wv & 1) * 2;       // 0 or 2
  const int ml = lane & 15;
  const int g = lane >> 4;

  v8f acc[2] = {};

  for (int k0 = 0; k0 < NDIM; k0 += 32) {
    __syncthreads();
    {
      const int row = tid >> 2;                 // 0..63
      const int koff = (tid & 3) * 8;           // 0,8,16,24
      const float* sq = qp + (size_t)row * NDIM + k0 + koff;
      const float* sv = vp + (size_t)row * NDIM + k0 + koff;
#pragma unroll
      for (int e = 0; e < 8; ++e) Qs[row][koff + e] = sq[e];
#pragma unroll
      for (int e = 0; e < 8; ++e) Vs[row][koff + e] = sv[e];
    }
    __syncthreads();

#pragma unroll
    for (int kk = 0; kk < 32; kk += 4) {
      v2f a;
      a.x = Qs[tm * 16 + ml][kk + 2 * g];
      a.y = Qs[tm * 16 + ml][kk + 2 * g + 1];
#pragma unroll
      for (int ni = 0; ni < 2; ++ni) {
        v2f bb;
        bb.x = Vs[(tn2 + ni) * 16 + ml][kk + 2 * g];      // B[k][n] = v[n][k]
        bb.y = Vs[(tn2 + ni) * 16 + ml][kk + 2 * g + 1];
        acc[ni] = WMMA_F32X4(a, bb, acc[ni]);
      }
    }
  }

#pragma unroll
  for (int ni = 0; ni < 2; ++ni) {
    const int ccol = (tn2 + ni) * 16 + ml;
#pragma unroll
    for (int e = 0; e < 8; ++e) {
      const int crow = tm * 16 + e + 8 * g;
      qvb[crow * HDIM + ccol] = acc[ni][e];
    }
  }
}

// ---------------------------------------------------------------------------
// attn[b,h] (64 x 1024) = qv[b,h] (64 x 64) @ ksoft[b,h] (64 x 1024)
// qv staged in LDS once; k streamed from global (L2-resident).
// Grid: (n tiles of 256, bh).  Wave w covers n offset w*32.
// ---------------------------------------------------------------------------
__global__ __launch_bounds__(256) void attn_wmma_kernel(const float* __restrict__ qkv,
                                                        const float* __restrict__ qvin,
                                                        float* __restrict__ attn) {
  const int n0 = blockIdx.x * 256;
  const int bh = blockIdx.y;
  const int b = bh >> 3;
  const int h = bh & 7;
  const float* __restrict__ kp = qkv + ((size_t)b * QKVROWS + CDIM + h * HDIM) * NDIM;
  const float* __restrict__ qvb = qvin + (size_t)bh * HDIM * HDIM;
  float* __restrict__ ob = attn + (size_t)bh * HDIM * NDIM;

  __shared__ alignas(16) float Qv[64][68];        // [c][d] padded

  const int tid = threadIdx.x;
  const int lane = tid & 31;
  const int wv = tid >> 5;
  const int ml = lane & 15;
  const int g = lane >> 4;

  {
    const int row = tid >> 2;                  // 0..63
    const int koff = (tid & 3) * 16;           // 0,16,32,48
    const float* src = qvb + row * HDIM + koff;
#pragma unroll
    for (int e = 0; e < 16; ++e) Qv[row][koff + e] = src[e];
  }
  __syncthreads();

  v8f acc[4][2] = {};

#pragma unroll
  for (int kk = 0; kk < HDIM; kk += 4) {
    v2f a[4];
#pragma unroll
    for (int mi = 0; mi < 4; ++mi) {
      a[mi].x = Qv[mi * 16 + ml][kk + 2 * g];
      a[mi].y = Qv[mi * 16 + ml][kk + 2 * g + 1];
    }
#pragma unroll
    for (int ni = 0; ni < 2; ++ni) {
      const int nc = n0 + wv * 32 + ni * 16 + ml;
      v2f bb;
      bb.x = kp[(size_t)(kk + 2 * g) * NDIM + nc];
      bb.y = kp[(size_t)(kk + 2 * g + 1) * NDIM + nc];
#pragma unroll
      for (int mi = 0; mi < 4; ++mi)
        acc[mi][ni] = WMMA_F32X4(a[mi], bb, acc[mi][ni]);
    }
  }

#pragma unroll
  for (int mi = 0; mi < 4; ++mi) {
#pragma unroll
    for (int ni = 0; ni < 2; ++ni) {
      const int nc = n0 + wv * 32 + ni * 16 + ml;
#pragma unroll
      for (int e = 0; e < 8; ++e) {
        const int crow = mi * 16 + e + 8 * g;
        ob[(size_t)crow * NDIM + nc] = acc[mi][ni][e];
      }
    }
  }
}

// ---------------------------------------------------------------------------
// Host-side launch
// ---------------------------------------------------------------------------
extern "C" void kernel_launch(void* const* d_in, const int* in_sizes, int n_in,
                              void* d_out, int out_size, void* d_ws, size_t ws_size,
                              hipStream_t stream) {
  const float* x = (const float*)d_in[0];       // [16, 512, 32, 32]
  const float* w_qkv = (const float*)d_in[1];   // [1536, 512]
  const float* w_proj = (const float*)d_in[2];  // [512, 512]
  const float* b_proj = (const float*)d_in[3];  // [512]
  float* out = (float*)d_out;                   // [16, 512, 32, 32]

  char* ws = (char*)d_ws;
  const size_t qkv_bytes = (size_t)BATCH * QKVROWS * NDIM * sizeof(float);     // 96 MB
  const size_t qv_bytes = (size_t)BATCH * HEADS * HDIM * HDIM * sizeof(float); // 2 MB
  float* qkv = (float*)ws;
  float* qv = (float*)(ws + qkv_bytes);
  float* attn = (float*)(ws + qkv_bytes + qv_bytes);                           // 32 MB

  const dim3 blk(256);

  // 1) qkv = w_qkv @ x     [16, 1536, 1024]
  gemm_wmma_f32_kernel<<<dim3(NDIM / 128, QKVROWS / 128, BATCH), blk, 0, stream>>>(
      w_qkv, x, nullptr, qkv, QKVROWS, CDIM, NDIM);

  // 2) softmax(k) over n; softmax(q) over head channels — in place
  softmax_k_kernel<<<dim3(BATCH * CDIM), blk, 0, stream>>>(qkv);
  softmax_q_kernel<<<dim3((BATCH * HEADS * NDIM) / 256), blk, 0, stream>>>(qkv);

  // 3) qv = q @ v^T per (b,h)
  qv_wmma_kernel<<<dim3(BATCH * HEADS), blk, 0, stream>>>(qkv, qv);

  // 4) attn = qv @ ksoft per (b,h)
  attn_wmma_kernel<<<dim3(NDIM / 256, BATCH * HEADS), blk, 0, stream>>>(qkv, qv, attn);

  // 5) out = w_proj @ attn + b_proj
  gemm_wmma_f32_kernel<<<dim3(NDIM / 128, CDIM / 128, BATCH), blk, 0, stream>>>(
      w_proj, attn, b_proj, out, CDIM, CDIM, NDIM);
}